// SpecialSpmm_15504831938955
// MI455X (gfx1250) — compile-verified
//
#include <hip/hip_runtime.h>
#include <stdint.h>

// ---------------------------------------------------------------------------
// COO SpMM: out[row[e], :] += values[e] * b[col[e], :]   (N=100k, E=3.2M, d=64)
// Bandwidth/atomic-bound; b and out are L2-resident (25.6MB each vs 192MB L2).
// Design: half-wave (16 lanes) per edge, float4 gather of b row (256B),
// global_atomic_add_f32 scatter (atomics resolve in L2). Edge arrays are
// streamed into LDS with CDNA5 async global->LDS copies, double-buffered.
// ---------------------------------------------------------------------------

#define BLOCK 256
#define CHUNK 256   // edges staged per chunk (== BLOCK, one element per thread)
#define D_FEAT 64

#if defined(__HIP_DEVICE_COMPILE__)
#if __has_builtin(__builtin_amdgcn_global_load_async_to_lds_b32) && \
    __has_builtin(__builtin_amdgcn_s_wait_asynccnt)
#define USE_ASYNC_LDS 1
#endif
#endif

#if defined(USE_ASYNC_LDS)
typedef __attribute__((address_space(1))) int as1_int;  // global
typedef __attribute__((address_space(3))) int as3_int;  // LDS
#define WAIT_ASYNC(n) __builtin_amdgcn_s_wait_asynccnt(n)
#else
#define WAIT_ASYNC(n)
#endif

// Stage one chunk of edge data (values/rows/cols) into LDS.
// Each of the 256 threads moves one element of each array.
// Async path: each wave issues exactly 3 async-to-LDS ops per chunk.
__device__ __forceinline__ void stage_chunk(const float* __restrict__ values,
                                            const int* __restrict__ rows,
                                            const int* __restrict__ cols,
                                            int base, int t,
                                            float* sval, int* srow, int* scol) {
#if defined(USE_ASYNC_LDS)
  __builtin_amdgcn_global_load_async_to_lds_b32(
      (as1_int*)(values + base + t), (as3_int*)(sval + t), 0, 0);
  __builtin_amdgcn_global_load_async_to_lds_b32(
      (as1_int*)(rows + base + t), (as3_int*)(srow + t), 0, 0);
  __builtin_amdgcn_global_load_async_to_lds_b32(
      (as1_int*)(cols + base + t), (as3_int*)(scol + t), 0, 0);
#else
  sval[t] = values[base + t];
  srow[t] = rows[base + t];
  scol[t] = cols[base + t];
#endif
}

__device__ __forceinline__ void scatter_edge(const float* __restrict__ b,
                                             float* __restrict__ out,
                                             float v, int r, int c, int lane16) {
  const float4 bv =
      reinterpret_cast<const float4*>(b + (size_t)c * D_FEAT)[lane16];
  float* o = out + (size_t)r * D_FEAT + (size_t)lane16 * 4u;
  atomicAdd(o + 0, v * bv.x);
  atomicAdd(o + 1, v * bv.y);
  atomicAdd(o + 2, v * bv.z);
  atomicAdd(o + 3, v * bv.w);
}

__global__ __launch_bounds__(BLOCK) void spmm_scatter_kernel(
    const float* __restrict__ values,
    const float* __restrict__ b,
    const int* __restrict__ rows,
    const int* __restrict__ cols,
    float* __restrict__ out,
    int n_edges, int n_full /* number of full CHUNK-sized chunks */) {

  __shared__ float s_val[2][CHUNK];
  __shared__ int s_row[2][CHUNK];
  __shared__ int s_col[2][CHUNK];

  const int t = threadIdx.x;
  const int lane16 = t & 15;
  const int egrp = t >> 4;  // 0..15 : which edge of the chunk-pass this half-wave owns

  int buf = 0;
  const int first = blockIdx.x;

  if (first < n_full) {
    stage_chunk(values, rows, cols, first * CHUNK, t, s_val[0], s_row[0], s_col[0]);
  }

  for (int chunk = first; chunk < n_full; chunk += gridDim.x) {
    const int nxt = chunk + gridDim.x;
    if (nxt < n_full) {
      stage_chunk(values, rows, cols, nxt * CHUNK, t,
                  s_val[buf ^ 1], s_row[buf ^ 1], s_col[buf ^ 1]);
      WAIT_ASYNC(3);  // retire the 3 older async ops (current buffer ready)
    } else {
      WAIT_ASYNC(0);
    }
    __syncthreads();

    // 256 edges per chunk; each half-wave walks 16 of them.
#pragma unroll 4
    for (int e = egrp; e < CHUNK; e += 16) {
      const float v = s_val[buf][e];
      const int r = s_row[buf][e];
      const int c = s_col[buf][e];
      scatter_edge(b, out, v, r, c, lane16);
    }
    __syncthreads();  // protect buf from being overwritten next iteration
    buf ^= 1;
  }

  // Remainder edges (n_edges % CHUNK), handled directly from global by block 0.
  if (blockIdx.x == 0) {
    for (int eg = n_full * CHUNK + egrp; eg < n_edges; eg += 16) {
      scatter_edge(b, out, values[eg], rows[eg], cols[eg], lane16);
    }
  }
}

__global__ __launch_bounds__(BLOCK) void spmm_zero_kernel(float4* __restrict__ out4,
                                                          int n4) {
  int i = blockIdx.x * blockDim.x + threadIdx.x;
  int stride = gridDim.x * blockDim.x;
  float4 z = make_float4(0.f, 0.f, 0.f, 0.f);
  for (; i < n4; i += stride) out4[i] = z;
}

extern "C" void kernel_launch(void* const* d_in, const int* in_sizes, int n_in,
                              void* d_out, int out_size, void* d_ws, size_t ws_size,
                              hipStream_t stream) {
  (void)d_ws; (void)ws_size; (void)n_in;

  const float* values = (const float*)d_in[0];
  const float* b      = (const float*)d_in[1];
  const int*   idx    = (const int*)d_in[2];   // [2, E] int32 (JAX x64-off)
  const int n_edges = in_sizes[0];
  const int* rows = idx;
  const int* cols = idx + n_edges;

  float* out = (float*)d_out;

  // 1) zero the output (atomics accumulate into it)
  {
    int n4 = out_size / 4;
    int grid = (n4 + BLOCK - 1) / BLOCK;
    if (grid > 8192) grid = 8192;
    if (grid < 1) grid = 1;
    spmm_zero_kernel<<<grid, BLOCK, 0, stream>>>((float4*)d_out, n4);
  }

  // 2) edge scatter: up to 6250 blocks -> exactly 2 chunks per block
  //    (minimum depth for the async double-buffer to overlap).
  {
    const int n_full = n_edges / CHUNK;
    int grid = n_full < 6250 ? n_full : 6250;
    if (grid < 1) grid = 1;
    spmm_scatter_kernel<<<grid, BLOCK, 0, stream>>>(values, b, rows, cols, out,
                                                    n_edges, n_full);
  }
}